// VectorQuantizerLayer_87179246174670
// MI455X (gfx1250) — compile-verified
//
#include <hip/hip_runtime.h>

typedef float v2f __attribute__((ext_vector_type(2)));
typedef float v8f __attribute__((ext_vector_type(8)));

#define D_DIM 64
#define K_CODES 512
#define N_ROWS (64*64*64)          // 262144 rows of 64
#define XST 68                     // padded LDS stride for staged x rows
#define WPB 8                      // waves per block
#define TPW 4                      // row tiles per wave
#define ROWS_PER_BLOCK (WPB*TPW*16)          // 512
#define NBLOCKS (N_ROWS/ROWS_PER_BLOCK)      // 512

// dynamic LDS layout (floats)
// Eswz: WMMA-fragment-swizzled codebook: chunk = (t*8+s2)*32+lane, 4 floats each
//       lane chunk = { E[8*s2+2h][k], E[8*s2+2h+1][k], E[8*s2+2h+4][k], E[8*s2+2h+5][k] },
//       h = lane>>4, k = t*16 + (lane&15)
#define SM_E    0
#define SM_E_FLOATS (32*8*32*4)              // 32768 (128 KB)
#define SM_NORM SM_E_FLOATS                  // 512 norms
#define SM_XS   (SM_NORM + K_CODES)
#define SM_FLOATS (SM_XS + WPB*16*XST)       // 32768+512+8704 = 41984 -> 164 KB

__device__ __forceinline__ int eswz_off(int d, int k) {
    // inverse of the swizzle: element (d,k) inside Eswz
    int s2 = d >> 3, so2 = (d >> 2) & 1, h = (d >> 1) & 1, od = d & 1;
    return (((k >> 4) * 8 + s2) * 32 + h * 16 + (k & 15)) * 4 + so2 * 2 + od;
}

__global__ void vq_init_kernel(float* ws) {
    if (threadIdx.x == 0) ws[0] = 0.f;
}

__global__ __launch_bounds__(256) void vq_main_kernel(
        const float* __restrict__ x, const float* __restrict__ emb,
        float* __restrict__ out, float* __restrict__ ws)
{
    extern __shared__ float sm[];
    float* Eswz = sm + SM_E;
    float* Nlds = sm + SM_NORM;

    const int tid  = threadIdx.x;
    const int wave = tid >> 5;
    const int lane = tid & 31;
    const int half = lane >> 4;
    const int l15  = lane & 15;
    float* xs = sm + SM_XS + wave * (16 * XST);

    // ---- fill codebook into LDS in WMMA fragment order ----
    // 8192 float4 chunks; 4 coalesced global b32 loads + 1 conflict-free b128 store each
    #pragma unroll 4
    for (int i = 0; i < 32; ++i) {
        int chunk = i * 256 + tid;
        int t  = chunk >> 8;
        int s2 = (chunk >> 5) & 7;
        int ln = chunk & 31;
        int k  = t * 16 + (ln & 15);
        int d0 = s2 * 8 + (ln >> 4) * 2;
        float4 v;
        v.x = emb[(d0    ) * K_CODES + k];
        v.y = emb[(d0 + 1) * K_CODES + k];
        v.z = emb[(d0 + 4) * K_CODES + k];
        v.w = emb[(d0 + 5) * K_CODES + k];
        *(float4*)(Eswz + chunk * 4) = v;
    }
    __syncthreads();
    // ---- column norms ||e_k||^2 from swizzled LDS (b128 reads) ----
    for (int k = tid; k < K_CODES; k += 256) {
        int t = k >> 4, l = k & 15;
        float s = 0.f;
        #pragma unroll
        for (int s2 = 0; s2 < 8; ++s2) {
            float4 a = *(const float4*)(Eswz + ((t * 8 + s2) * 32 + l) * 4);
            float4 b = *(const float4*)(Eswz + ((t * 8 + s2) * 32 + 16 + l) * 4);
            s += a.x*a.x + a.y*a.y + a.z*a.z + a.w*a.w;
            s += b.x*b.x + b.y*b.y + b.z*b.z + b.w*b.w;
        }
        Nlds[k] = s;
    }
    __syncthreads();

    const float* bbase = Eswz + lane * 4;   // per-lane B-fragment base

    // ---- each wave processes TPW row tiles, reusing the LDS codebook ----
    for (int it = 0; it < TPW; ++it) {
        const int row0 = (blockIdx.x * (WPB * TPW) + wave * TPW + it) * 16;

        // stage this tile's 16 x-rows (4 KB contiguous) into LDS
        #pragma unroll
        for (int i = 0; i < 8; ++i) {
            int r = i * 2 + half;
            *(float4*)(xs + r * XST + l15 * 4) =
                *(const float4*)(x + (size_t)(row0 + r) * D_DIM + l15 * 4);
        }
        // A fragments (ISA layout: lanes 0-15 K=4s,4s+1 ; lanes 16-31 K=4s+2,4s+3)
        v2f A[16];
        #pragma unroll
        for (int s = 0; s < 16; ++s)
            A[s] = *(const v2f*)(xs + l15 * XST + s * 4 + 2 * half);

        float bestv[8];
        int   besti[8];
        #pragma unroll
        for (int j = 0; j < 8; ++j) { bestv[j] = 3.4e38f; besti[j] = 0; }

        for (int t = 0; t < 32; t += 2) {
            const float* bt = bbase + t * 1024;   // tile t fragments
            v8f acc0 = {};
            v8f acc1 = {};
            #pragma unroll
            for (int s2 = 0; s2 < 8; ++s2) {
                float4 B0 = *(const float4*)(bt + s2 * 128);          // tile t
                float4 B1 = *(const float4*)(bt + 1024 + s2 * 128);   // tile t+1
                v2f b;
                b.x = B0.x; b.y = B0.y;
                acc0 = __builtin_amdgcn_wmma_f32_16x16x4_f32(
                    false, A[2*s2], false, b, (short)0, acc0, false, false);
                b.x = B1.x; b.y = B1.y;
                acc1 = __builtin_amdgcn_wmma_f32_16x16x4_f32(
                    false, A[2*s2], false, b, (short)0, acc1, false, false);
                b.x = B0.z; b.y = B0.w;
                acc0 = __builtin_amdgcn_wmma_f32_16x16x4_f32(
                    false, A[2*s2+1], false, b, (short)0, acc0, false, false);
                b.x = B1.z; b.y = B1.w;
                acc1 = __builtin_amdgcn_wmma_f32_16x16x4_f32(
                    false, A[2*s2+1], false, b, (short)0, acc1, false, false);
            }
            float n0 = Nlds[t * 16 + l15];
            float n1 = Nlds[t * 16 + 16 + l15];
            int   c0 = t * 16 + l15, c1 = c0 + 16;
            #pragma unroll
            for (int j = 0; j < 8; ++j) {
                float s0 = n0 - 2.f * acc0[j];
                float s1 = n1 - 2.f * acc1[j];
                if (s0 < bestv[j]) { bestv[j] = s0; besti[j] = c0; }
                if (s1 < bestv[j]) { bestv[j] = s1; besti[j] = c1; }
            }
        }

        // butterfly min-reduce across the 16 lanes of each half
        #pragma unroll
        for (int j = 0; j < 8; ++j) {
            float v  = bestv[j];
            int   ix = besti[j];
            #pragma unroll
            for (int m = 1; m < 16; m <<= 1) {
                float ov = __shfl_xor(v, m, 32);
                int   oi = __shfl_xor(ix, m, 32);
                if (ov < v || (ov == v && oi < ix)) { v = ov; ix = oi; }
            }
            bestv[j] = v; besti[j] = ix;
        }

        // epilogue: gather code rows, write quantized output, accumulate SSE
        float sse = 0.f;
        #pragma unroll
        for (int j = 0; j < 8; ++j) {
            int r = j + 8 * half;
            int k = besti[j];
            float* orow = out + (size_t)(row0 + r) * D_DIM;
            #pragma unroll
            for (int c = 0; c < 4; ++c) {
                int d = c * 16 + l15;
                float q  = Eswz[eswz_off(d, k)];
                float xv = xs[r * XST + d];
                float df = q - xv;
                sse += df * df;
                orow[d] = q;
            }
        }
        #pragma unroll
        for (int m = 1; m < 32; m <<= 1) sse += __shfl_xor(sse, m, 32);
        if (lane == 0) atomicAdd(ws, sse);
    }
}

__global__ void vq_fin_kernel(const float* __restrict__ ws, float* __restrict__ out) {
    // loss = quantizer + BETA*commitment = (1 + 0.25) * mean((q-x)^2)
    out[(size_t)N_ROWS * D_DIM] = 1.25f * ws[0] * (1.0f / 16777216.0f);
}

extern "C" void kernel_launch(void* const* d_in, const int* in_sizes, int n_in,
                              void* d_out, int out_size, void* d_ws, size_t ws_size,
                              hipStream_t stream) {
    const float* x   = (const float*)d_in[0];   // [64,64,64,64] f32
    const float* emb = (const float*)d_in[1];   // [64,512] f32
    float* out = (float*)d_out;                 // 64^4 quantized + 1 loss
    float* ws  = (float*)d_ws;

    vq_init_kernel<<<1, 32, 0, stream>>>(ws);
    size_t smem = (size_t)SM_FLOATS * sizeof(float);
    vq_main_kernel<<<NBLOCKS, 256, smem, stream>>>(x, emb, out, ws);
    vq_fin_kernel<<<1, 1, 0, stream>>>(ws, out);
}